// MoEBudgetAwareINLLayer_40578851012929
// MI455X (gfx1250) — compile-verified
//
#include <hip/hip_runtime.h>

// MoEBudgetAwareINLLayer forward, degenerate (no MoE controller) case:
//   repeat n times:  v = 0.5*v - 0.1*(x - mu);  x = x + 0.9*v
// Purely elementwise over (B,D)=(8192,2048) fp32. h is unused.
// Memory-bound: 320 MB streamed once -> NT-hinted b128 loads/stores, wave32.

typedef float v4f __attribute__((ext_vector_type(4)));

#define ALPHA 0.5f
#define BETA  0.1f
#define GATE  0.9f

__global__ __launch_bounds__(256) void inl_step_vec4(
    const v4f* __restrict__ x0,
    const v4f* __restrict__ v0,
    const v4f* __restrict__ mu,
    const int* __restrict__ n_iter,   // uniform: scalar load + s_wait_kmcnt
    v4f* __restrict__ x_out,
    v4f* __restrict__ v_out,
    long long n4)
{
    const int n = *n_iter;  // same address for all lanes -> SGPR
    long long i = (long long)blockIdx.x * blockDim.x + threadIdx.x;
    if (i >= n4) return;

    // Streaming data, zero reuse: non-temporal hints (TH=NT) keep the
    // 192MB L2 clean and avoid write-allocate rinse on the stores.
    v4f x = __builtin_nontemporal_load(&x0[i]);
    v4f v = __builtin_nontemporal_load(&v0[i]);
    v4f m = __builtin_nontemporal_load(&mu[i]);

    for (int k = 0; k < n; ++k) {
        v4f err = x - m;
        v = ALPHA * v - BETA * err;   // contracts to v_fma
        x = x + GATE * v;
    }

    __builtin_nontemporal_store(x, &x_out[i]);
    __builtin_nontemporal_store(v, &v_out[i]);
}

// Scalar tail for element counts not divisible by 4 (never fires for 8192x2048,
// kept for generality; launch decision depends only on in_sizes -> deterministic).
__global__ __launch_bounds__(64) void inl_step_tail(
    const float* __restrict__ x0,
    const float* __restrict__ v0,
    const float* __restrict__ mu,
    const int* __restrict__ n_iter,
    float* __restrict__ x_out,
    float* __restrict__ v_out,
    long long base, long long total)
{
    const int n = *n_iter;
    long long i = base + (long long)blockIdx.x * blockDim.x + threadIdx.x;
    if (i >= total) return;
    float x = x0[i], v = v0[i], m = mu[i];
    for (int k = 0; k < n; ++k) {
        float err = x - m;
        v = ALPHA * v - BETA * err;
        x = x + GATE * v;
    }
    __builtin_nontemporal_store(x, &x_out[i]);
    __builtin_nontemporal_store(v, &v_out[i]);
}

extern "C" void kernel_launch(void* const* d_in, const int* in_sizes, int n_in,
                              void* d_out, int out_size, void* d_ws, size_t ws_size,
                              hipStream_t stream) {
    // setup_inputs order: 0:h (unused), 1:x_init, 2:v_init, 3:mu, 4:default_iterations
    const float* x0  = (const float*)d_in[1];
    const float* v0  = (const float*)d_in[2];
    const float* mu  = (const float*)d_in[3];
    const int*   nit = (const int*)d_in[4];

    const long long n = (long long)in_sizes[1];   // B*D elements
    float* x_out = (float*)d_out;                 // outputs concatenated: x then v
    float* v_out = x_out + n;

    const long long n4  = n >> 2;                 // float4 elements
    const long long rem = n - (n4 << 2);

    if (n4 > 0) {
        const int threads = 256;
        const long long blocks = (n4 + threads - 1) / threads;
        inl_step_vec4<<<(unsigned)blocks, threads, 0, stream>>>(
            (const v4f*)x0, (const v4f*)v0, (const v4f*)mu, nit,
            (v4f*)x_out, (v4f*)v_out, n4);
    }
    if (rem > 0) {
        inl_step_tail<<<1, 64, 0, stream>>>(
            x0, v0, mu, nit, x_out, v_out, n4 << 2, n);
    }
}